// Conv_12962211300035
// MI455X (gfx1250) — compile-verified
//
#include <hip/hip_runtime.h>
#include <hip/hip_bf16.h>
#include <math.h>

#define MULW 16
#define NNODES 12500
#define NEDGES 200000
#define TILE 16
#define NTILES (NEDGES / TILE)   // 12500
#define WPB 4                    // waves (edge tiles) per block
#define BLOCK (WPB * 32)

typedef float v2f __attribute__((ext_vector_type(2)));
typedef float v8f __attribute__((ext_vector_type(8)));

__device__ __forceinline__ float silu_f(float x) {
    return x / (1.0f + __expf(-x));
}

struct WaveLds {
    float hS[16][17];        // h tile, padded vs bank conflicts
    float eaS[16][17];       // edge_attr tile
    float c0T[16][16];       // [u][e]  coeff for slot 0 -> out0
    float c1T[16][16];       // [u][e]  coeff for slot 1 -> out0
    float c2T[3][16][16];    // [i][u][e] coeff for slot 2 -> out1[i]
    float c3T[3][16][16];    // [i][u][e] coeff for slot 3 -> out1[i]
    int   dstS[16];
};

__global__ __launch_bounds__(BLOCK) void conv_edge_kernel(
    const float* __restrict__ node_attr,   // [NNODES,64]
    const int*   __restrict__ edge_index,  // [2,NEDGES]
    const float* __restrict__ edge_attr,   // [NEDGES,16]
    const float* __restrict__ edge_sh,     // [NEDGES,4]
    const float* __restrict__ w1,          // [16,16]
    const float* __restrict__ b1,          // [16]
    const float* __restrict__ w2,          // [16,1024]
    const float* __restrict__ b2,          // [1024]
    float* __restrict__ acc,               // [NNODES*64]
    float* __restrict__ counts)            // [NNODES]
{
    // w2 pre-swizzled into WMMA B-fragment order:
    //   w2f[((ku)*4 + q)*32 + lane] = { w2[4q+2*hi][ku*16+n], w2[4q+2*hi+1][ku*16+n] }
    // One ds_load_b64 per fragment, conflict-free (lane stride = 2 dwords -> 64 banks).
    __shared__ v2f   w2f[64 * 4 * 32];   // 64 KB
    __shared__ float b2s[1024];
    __shared__ float w1s[16][17];
    __shared__ float b1s[16];
    __shared__ WaveLds wl[WPB];

    const int tid  = threadIdx.x;
    const int wave = tid >> 5;
    const int lane = tid & 31;
    const int n    = lane & 15;   // column / edge-in-half index
    const int hi   = lane >> 4;   // which half of the wave

    // ---- cooperative staging of shared weights (permuted into fragment order) ----
    for (int i = tid; i < 64 * 4 * 32; i += BLOCK) {
        const int l  = i & 31;
        const int q  = (i >> 5) & 3;
        const int ku = i >> 7;
        const int fr  = 4 * q + 2 * (l >> 4);
        const int col = ku * 16 + (l & 15);
        v2f v;
        v.x = w2[fr * 1024 + col];
        v.y = w2[(fr + 1) * 1024 + col];
        w2f[i] = v;
    }
    for (int i = tid; i < 256; i += BLOCK)
        ((float4*)b2s)[i] = ((const float4*)b2)[i];
    for (int i = tid; i < 256; i += BLOCK)
        w1s[i >> 4][i & 15] = w1[i];
    if (tid < 16) b1s[tid] = b1[tid];

    const int tile   = blockIdx.x * WPB + wave;  // grid sized so tile < NTILES always
    const int e_base = tile * TILE;
    WaveLds& L = wl[wave];

    // ---- stage edge_attr tile (16x16) ----
    for (int j = lane; j < 64; j += 32) {
        const int r = j >> 2, c4 = j & 3;
        float4 v = ((const float4*)(edge_attr + (size_t)(e_base + r) * 16))[c4];
        L.eaS[r][c4 * 4 + 0] = v.x;
        L.eaS[r][c4 * 4 + 1] = v.y;
        L.eaS[r][c4 * 4 + 2] = v.z;
        L.eaS[r][c4 * 4 + 3] = v.w;
    }
    if (hi == 0) L.dstS[n] = edge_index[NEDGES + e_base + n];

    // ---- gather node features, build coefficient tables ----
    {
        const float4 sh = *(const float4*)(edge_sh + (size_t)(e_base + n) * 4);
        const float y0 = sh.x, y1x = sh.y, y1y = sh.z, y1z = sh.w;
        const int   se = edge_index[e_base + n];
        const float* xr = node_attr + (size_t)se * 64;
        const float inv_sqrt3 = 0.57735026918962576f;
        #pragma unroll
        for (int uu = 0; uu < 8; ++uu) {
            const int u = hi * 8 + uu;
            const float x0u = xr[u];
            const float x10 = xr[16 + u * 3 + 0];
            const float x11 = xr[16 + u * 3 + 1];
            const float x12 = xr[16 + u * 3 + 2];
            L.c0T[u][n]    = x0u * y0;
            L.c1T[u][n]    = inv_sqrt3 * (x10 * y1x + x11 * y1y + x12 * y1z);
            L.c2T[0][u][n] = x0u * y1x;
            L.c2T[1][u][n] = x0u * y1y;
            L.c2T[2][u][n] = x0u * y1z;
            L.c3T[0][u][n] = x10 * y0;
            L.c3T[1][u][n] = x11 * y0;
            L.c3T[2][u][n] = x12 * y0;
        }
    }
    __syncthreads();   // weights staged + tile LDS visible

    // ---- stage 1 MLP: h = silu(ea @ w1 + b1); lane does e=n, f in [hi*8, hi*8+8) ----
    #pragma unroll
    for (int ff = 0; ff < 8; ++ff) {
        const int f = hi * 8 + ff;
        float a = b1s[f];
        #pragma unroll
        for (int d = 0; d < 16; ++d) a = fmaf(L.eaS[n][d], w1s[d][f], a);
        L.hS[n][f] = silu_f(a);
    }

    // ---- A fragments for h (16 edges x 16 features), K chained in chunks of 4 ----
    v2f afrag[4];
    #pragma unroll
    for (int q = 0; q < 4; ++q) {
        const int fr = 4 * q + 2 * hi;
        v2f a;
        a.x = L.hS[n][fr];
        a.y = L.hS[n][fr + 1];
        afrag[q] = a;
    }

    float out0[8], out1a[8], out1b[8], out1c[8];
    #pragma unroll
    for (int r = 0; r < 8; ++r) { out0[r] = 0.f; out1a[r] = 0.f; out1b[r] = 0.f; out1c[r] = 0.f; }

    const int e8 = hi * 8;   // first edge-row this lane accumulates

    for (int u = 0; u < 16; ++u) {
        // ---------- slot 0: out0 += c0 * (h@w2 + b2)[k=0,u] ----------
        {
            const int ku = 0 * 16 + u;
            v8f d = {0.f, 0.f, 0.f, 0.f, 0.f, 0.f, 0.f, 0.f};
            #pragma unroll
            for (int q = 0; q < 4; ++q) {
                const v2f b = w2f[(ku * 4 + q) * 32 + lane];
                d = __builtin_amdgcn_wmma_f32_16x16x4_f32(false, afrag[q], false, b,
                                                          (short)0, d, false, false);
            }
            const float bb = b2s[ku * 16 + n];
            const float4 cA = *(const float4*)&L.c0T[u][e8];
            const float4 cB = *(const float4*)&L.c0T[u][e8 + 4];
            const float cv[8] = {cA.x, cA.y, cA.z, cA.w, cB.x, cB.y, cB.z, cB.w};
            #pragma unroll
            for (int r = 0; r < 8; ++r) out0[r] = fmaf(cv[r], d[r] + bb, out0[r]);
        }
        // ---------- slot 1: out0 += c1 * W[k=1,u] ----------
        {
            const int ku = 1 * 16 + u;
            v8f d = {0.f, 0.f, 0.f, 0.f, 0.f, 0.f, 0.f, 0.f};
            #pragma unroll
            for (int q = 0; q < 4; ++q) {
                const v2f b = w2f[(ku * 4 + q) * 32 + lane];
                d = __builtin_amdgcn_wmma_f32_16x16x4_f32(false, afrag[q], false, b,
                                                          (short)0, d, false, false);
            }
            const float bb = b2s[ku * 16 + n];
            const float4 cA = *(const float4*)&L.c1T[u][e8];
            const float4 cB = *(const float4*)&L.c1T[u][e8 + 4];
            const float cv[8] = {cA.x, cA.y, cA.z, cA.w, cB.x, cB.y, cB.z, cB.w};
            #pragma unroll
            for (int r = 0; r < 8; ++r) out0[r] = fmaf(cv[r], d[r] + bb, out0[r]);
        }
        // ---------- slot 2: out1[i] += c2[i] * W[k=2,u] ----------
        {
            const int ku = 2 * 16 + u;
            v8f d = {0.f, 0.f, 0.f, 0.f, 0.f, 0.f, 0.f, 0.f};
            #pragma unroll
            for (int q = 0; q < 4; ++q) {
                const v2f b = w2f[(ku * 4 + q) * 32 + lane];
                d = __builtin_amdgcn_wmma_f32_16x16x4_f32(false, afrag[q], false, b,
                                                          (short)0, d, false, false);
            }
            const float bb = b2s[ku * 16 + n];
            #pragma unroll
            for (int i = 0; i < 3; ++i) {
                const float4 cA = *(const float4*)&L.c2T[i][u][e8];
                const float4 cB = *(const float4*)&L.c2T[i][u][e8 + 4];
                const float cv[8] = {cA.x, cA.y, cA.z, cA.w, cB.x, cB.y, cB.z, cB.w};
                float* o = (i == 0) ? out1a : (i == 1) ? out1b : out1c;
                #pragma unroll
                for (int r = 0; r < 8; ++r) o[r] = fmaf(cv[r], d[r] + bb, o[r]);
            }
        }
        // ---------- slot 3: out1[i] += c3[i] * W[k=3,u] ----------
        {
            const int ku = 3 * 16 + u;
            v8f d = {0.f, 0.f, 0.f, 0.f, 0.f, 0.f, 0.f, 0.f};
            #pragma unroll
            for (int q = 0; q < 4; ++q) {
                const v2f b = w2f[(ku * 4 + q) * 32 + lane];
                d = __builtin_amdgcn_wmma_f32_16x16x4_f32(false, afrag[q], false, b,
                                                          (short)0, d, false, false);
            }
            const float bb = b2s[ku * 16 + n];
            #pragma unroll
            for (int i = 0; i < 3; ++i) {
                const float4 cA = *(const float4*)&L.c3T[i][u][e8];
                const float4 cB = *(const float4*)&L.c3T[i][u][e8 + 4];
                const float cv[8] = {cA.x, cA.y, cA.z, cA.w, cB.x, cB.y, cB.z, cB.w};
                float* o = (i == 0) ? out1a : (i == 1) ? out1b : out1c;
                #pragma unroll
                for (int r = 0; r < 8; ++r) o[r] = fmaf(cv[r], d[r] + bb, o[r]);
            }
        }
    }

    // ---- scatter: atomics into per-node accumulator ----
    const float alpha = 0.17677669529663687f;  // 1/sqrt(2*MUL)
    int dsts[8];
    #pragma unroll
    for (int r = 0; r < 8; ++r) dsts[r] = L.dstS[e8 + r];
    #pragma unroll
    for (int r = 0; r < 8; ++r) {
        float* base = acc + (size_t)dsts[r] * 64;
        __hip_atomic_fetch_add(base + n, alpha * out0[r],
                               __ATOMIC_RELAXED, __HIP_MEMORY_SCOPE_AGENT);
        __hip_atomic_fetch_add(base + 16 + n * 3 + 0, alpha * out1a[r],
                               __ATOMIC_RELAXED, __HIP_MEMORY_SCOPE_AGENT);
        __hip_atomic_fetch_add(base + 16 + n * 3 + 1, alpha * out1b[r],
                               __ATOMIC_RELAXED, __HIP_MEMORY_SCOPE_AGENT);
        __hip_atomic_fetch_add(base + 16 + n * 3 + 2, alpha * out1c[r],
                               __ATOMIC_RELAXED, __HIP_MEMORY_SCOPE_AGENT);
    }
    if (hi == 0)
        __hip_atomic_fetch_add(counts + L.dstS[n], 1.0f,
                               __ATOMIC_RELAXED, __HIP_MEMORY_SCOPE_AGENT);
}

__global__ __launch_bounds__(256) void finalize_kernel(
    const float* __restrict__ acc, const float* __restrict__ counts,
    float* __restrict__ out)
{
    const int i = blockIdx.x * 256 + threadIdx.x;
    if (i < NNODES * 64) {
        out[i] = acc[i] / fmaxf(counts[i >> 6], 1.0f);
    }
}

extern "C" void kernel_launch(void* const* d_in, const int* in_sizes, int n_in,
                              void* d_out, int out_size, void* d_ws, size_t ws_size,
                              hipStream_t stream) {
    (void)in_sizes; (void)n_in; (void)out_size; (void)ws_size;
    const float* node_attr  = (const float*)d_in[0];
    const int*   edge_index = (const int*)  d_in[1];
    const float* edge_attr  = (const float*)d_in[2];
    const float* edge_sh    = (const float*)d_in[3];
    const float* w1         = (const float*)d_in[4];
    const float* b1         = (const float*)d_in[5];
    const float* w2         = (const float*)d_in[6];
    const float* b2         = (const float*)d_in[7];

    float* acc    = (float*)d_ws;              // NNODES*64 sums
    float* counts = acc + (size_t)NNODES * 64; // NNODES counts

    hipMemsetAsync(d_ws, 0, ((size_t)NNODES * 64 + NNODES) * sizeof(float), stream);

    const int nblocks = NTILES / WPB;  // 12500/4 = 3125, exact
    conv_edge_kernel<<<nblocks, BLOCK, 0, stream>>>(
        node_attr, edge_index, edge_attr, edge_sh, w1, b1, w2, b2, acc, counts);

    finalize_kernel<<<(NNODES * 64 + 255) / 256, 256, 0, stream>>>(acc, counts, (float*)d_out);
}